// Fusion_76089640616500
// MI455X (gfx1250) — compile-verified
//
#include <hip/hip_runtime.h>
#include <hip/hip_bf16.h>

typedef __attribute__((ext_vector_type(2))) float v2f;
typedef __attribute__((ext_vector_type(4))) float v4f;
typedef __attribute__((ext_vector_type(8))) float v8f;

#define HW_   256
#define HW2_  65536        // 256*256
#define C_    64
#define B_    8
#define NPIX_ 524288       // B*HW2

// ---------------------------------------------------------------------------
// Phase 1: channel L1 reduction via V_WMMA_F32_16X16X4_F32.
// Each wave owns 16 consecutive pixels of one image. A-matrix = |x| for
// 16 pixels x 4 channels (fp32), B-matrix = ones, f32 accumulator carries the
// sum across 16 chunks of 4 channels (C=64 total). Every column of D holds the
// per-pixel channel sum; lanes 0/16 write the 16 sums out.
// ---------------------------------------------------------------------------
__global__ __launch_bounds__(256) void l1_reduce_wmma(
    const float* __restrict__ x1, const float* __restrict__ x2,
    float* __restrict__ n1, float* __restrict__ n2)
{
    const int lane = threadIdx.x & 31;
    const int wave = threadIdx.x >> 5;
    const int group = blockIdx.x * 8 + wave;      // 16-pixel group id
    const int P    = group << 4;                  // global flat pixel index
    const int b    = P >> 16;                     // P / 65536
    const int pix  = P & (HW2_ - 1);
    const int half = lane >> 4;                   // 0: K=0,1  1: K=2,3
    const int m    = lane & 15;                   // A-matrix row = pixel

    const size_t imgBase = (size_t)b * C_ * HW2_ + (size_t)pix + (size_t)m;
    const float* p1 = x1 + imgBase;
    const float* p2 = x2 + imgBase;

    v2f ones; ones.x = 1.0f; ones.y = 1.0f;      // B = all ones (layout-free)
    v8f acc1 = {};
    v8f acc2 = {};

#pragma unroll
    for (int k = 0; k < 16; ++k) {
        const int c0 = 4 * k + 2 * half;
        v2f a1, a2;
        a1.x = __builtin_fabsf(__builtin_nontemporal_load(p1 + (size_t)c0       * HW2_));
        a1.y = __builtin_fabsf(__builtin_nontemporal_load(p1 + (size_t)(c0 + 1) * HW2_));
        a2.x = __builtin_fabsf(__builtin_nontemporal_load(p2 + (size_t)c0       * HW2_));
        a2.y = __builtin_fabsf(__builtin_nontemporal_load(p2 + (size_t)(c0 + 1) * HW2_));
        // D = A x ones + C  => every column of D = per-pixel partial L1 sum
        acc1 = __builtin_amdgcn_wmma_f32_16x16x4_f32(
            /*neg_a=*/false, a1, /*neg_b=*/false, ones,
            /*c_mod=*/(short)0, acc1, /*reuse_a=*/false, /*reuse_b=*/false);
        acc2 = __builtin_amdgcn_wmma_f32_16x16x4_f32(
            false, a2, false, ones, (short)0, acc2, false, false);
    }

    // D VGPR r: lanes 0-15 hold M=r, lanes 16-31 hold M=r+8 (all N identical).
    if (m == 0) {
        const int pbase = b * HW2_ + pix + half * 8;
#pragma unroll
        for (int r = 0; r < 8; ++r) {
            n1[pbase + r] = acc1[r];
            n2[pbase + r] = acc2[r];
        }
    }
}

// ---------------------------------------------------------------------------
// Phase 2: 3x3 conv (zero pad) + bias on the two 8x256x256 maps, then the
// ratio r_i = c_i / (c1 + c2). Tiny (1M pixels total) -> latency irrelevant.
// ---------------------------------------------------------------------------
__global__ __launch_bounds__(256) void conv_ratio(
    const float* __restrict__ n1, const float* __restrict__ n2,
    const float* __restrict__ w9, const float* __restrict__ bias,
    float* __restrict__ r1, float* __restrict__ r2)
{
    const int idx = blockIdx.x * 256 + threadIdx.x;   // 0 .. NPIX-1
    const int b   = idx >> 16;
    const int pix = idx & (HW2_ - 1);
    const int h   = pix >> 8;
    const int wv  = pix & (HW_ - 1);

    const float* base1 = n1 + b * HW2_;
    const float* base2 = n2 + b * HW2_;

    float c1 = 0.0f, c2 = 0.0f;
#pragma unroll
    for (int dy = -1; dy <= 1; ++dy) {
#pragma unroll
        for (int dx = -1; dx <= 1; ++dx) {
            const int hh = h + dy;
            const int ww = wv + dx;
            if (hh >= 0 && hh < HW_ && ww >= 0 && ww < HW_) {
                const float wt = w9[(dy + 1) * 3 + (dx + 1)];
                const int q = hh * HW_ + ww;
                c1 += wt * base1[q];
                c2 += wt * base2[q];
            }
        }
    }
    const float bb = bias[0];
    c1 += bb;
    c2 += bb;
    const float denom = c1 + c2;
    r1[idx] = c1 / denom;
    r2[idx] = c2 / denom;
}

// ---------------------------------------------------------------------------
// Phase 3: out = x1 * r1 + x2 * r2 (ratios broadcast over 64 channels).
// float4 streaming loads/stores for x1/x2/out; r maps (2 MB each) stay hot in
// L2 and are read through the normal cached path (64x reuse per element).
// ---------------------------------------------------------------------------
__global__ __launch_bounds__(256) void combine(
    const float* __restrict__ x1, const float* __restrict__ x2,
    const float* __restrict__ r1, const float* __restrict__ r2,
    float* __restrict__ out)
{
    const size_t t  = (size_t)blockIdx.x * 256 + threadIdx.x;
    const size_t i4 = t << 2;                       // element index (x4 vector)
    const size_t bc  = i4 >> 16;                    // b*C + c
    const int    pix = (int)(i4 & (HW2_ - 1));
    const size_t b   = bc >> 6;                     // / 64
    const size_t rIdx = (b << 16) + (size_t)pix;

    v4f a  = __builtin_nontemporal_load((const v4f*)(x1 + i4));
    v4f c  = __builtin_nontemporal_load((const v4f*)(x2 + i4));
    v4f ra = *(const v4f*)(r1 + rIdx);
    v4f rc = *(const v4f*)(r2 + rIdx);
    v4f o  = a * ra + c * rc;
    __builtin_nontemporal_store(o, (v4f*)(out + i4));
}

// ---------------------------------------------------------------------------
extern "C" void kernel_launch(void* const* d_in, const int* in_sizes, int n_in,
                              void* d_out, int out_size, void* d_ws, size_t ws_size,
                              hipStream_t stream)
{
    const float* x1   = (const float*)d_in[0];
    const float* x2   = (const float*)d_in[1];
    const float* w9   = (const float*)d_in[2];   // 9 floats (3x3 kernel)
    const float* bias = (const float*)d_in[3];   // 1 float

    float* out = (float*)d_out;
    float* ws  = (float*)d_ws;                   // 4 * NPIX_ floats = 8 MB
    float* n1  = ws;
    float* n2  = ws + (size_t)NPIX_;
    float* r1  = ws + (size_t)2 * NPIX_;
    float* r2  = ws + (size_t)3 * NPIX_;

    // Phase 1: 524288 pixels / 16 per wave / 8 waves per block = 4096 blocks
    l1_reduce_wmma<<<4096, 256, 0, stream>>>(x1, x2, n1, n2);

    // Phase 2: 524288 pixels / 256 = 2048 blocks
    conv_ratio<<<2048, 256, 0, stream>>>(n1, n2, w9, bias, r1, r2);

    // Phase 3: 33554432 elements / 4 per thread / 256 = 32768 blocks
    combine<<<32768, 256, 0, stream>>>(x1, x2, r1, r2, out);
}